// FineGrained_feat_83133386981701
// MI455X (gfx1250) — compile-verified
//
#include <hip/hip_runtime.h>
#include <stdint.h>

#define B 4
#define CCH 64
#define H 256
#define W 256
#define N (H * W)          // 65536
#define NCLS 4
#define HALFS 512
#define PAIRS 12           // B * (NCLS-1)
#define OUT_FEAT (PAIRS * CCH * 1024)  // 786432

typedef __attribute__((address_space(3))) unsigned char lds_u8_t;

// ---------------- Kernel 1: argmax over class dim (first-max wins) ----------------
__global__ void argmax_kernel(const float* __restrict__ preds, uint8_t* __restrict__ am) {
    int i = blockIdx.x * blockDim.x + threadIdx.x;   // [0, B*N)
    if (i >= B * N) return;
    int b = i >> 16;          // N == 65536
    int n = i & (N - 1);
    const float* p = preds + (size_t)b * NCLS * N + n;
    float best = p[0];
    int bi = 0;
#pragma unroll
    for (int c = 1; c < NCLS; ++c) {
        float v = p[(size_t)c * N];
        if (v > best) { best = v; bi = c; }   // strict '>' keeps first max (jnp.argmax)
    }
    am[i] = (uint8_t)bi;
}

// ---------------- Kernel 2: 3x3 max dilation via async global->LDS tile ----------------
// clamp-replicate == -inf SAME padding for a max-pool (duplicated edge value is
// already inside the window).
__global__ void dilate_kernel(const uint8_t* __restrict__ am, uint8_t* __restrict__ dil) {
    __shared__ uint8_t tile[18 * 18];
    int b   = blockIdx.z;
    int bx0 = blockIdx.x * 16, by0 = blockIdx.y * 16;
    int tid = threadIdx.y * 16 + threadIdx.x;
    const uint8_t* src = am + (size_t)b * N;

    uint32_t lds_base = (uint32_t)(uintptr_t)(lds_u8_t*)&tile[0];

    // Stage the 18x18 halo tile with CDNA5 async global->LDS byte copies.
    for (int k = tid; k < 18 * 18; k += 256) {
        int ty = by0 + (k / 18) - 1;
        int tx = bx0 + (k % 18) - 1;
        ty = min(max(ty, 0), H - 1);
        tx = min(max(tx, 0), W - 1);
        uint64_t ga = (uint64_t)(uintptr_t)(src + ty * W + tx);
        uint32_t la = lds_base + (uint32_t)k;
        asm volatile("global_load_async_to_lds_b8 %0, %1, off"
                     :: "v"(la), "v"(ga) : "memory");
    }
    asm volatile("s_wait_asynccnt 0" ::: "memory");
    __syncthreads();

    int lx = threadIdx.x, ly = threadIdx.y;
    uint8_t m = 0;
#pragma unroll
    for (int dy = 0; dy < 3; ++dy)
#pragma unroll
        for (int dx = 0; dx < 3; ++dx) {
            uint8_t v = tile[(ly + dy) * 18 + (lx + dx)];
            m = v > m ? v : m;
        }
    dil[(size_t)b * N + (by0 + ly) * W + (bx0 + lx)] = m;
}

// ---------------- Kernel 3: stable hard/easy selection scan (one block per pair) ----------------
__global__ void __launch_bounds__(1024)
select_kernel(const int* __restrict__ labels, const uint8_t* __restrict__ dil,
              int* __restrict__ sel) {
    int p = blockIdx.x;               // 0..11
    int b = p / 3, cls = p % 3 + 1;
    const int*     lab = labels + (size_t)b * N;
    const uint8_t* pd  = dil    + (size_t)b * N;
    int* s = sel + p * 1024;

    int tid  = threadIdx.x;           // 0..1023
    int lane = tid & 31, wid = tid >> 5;

    __shared__ int ssel[1024];
    __shared__ int wsH[32], wsE[32];
    __shared__ int sBaseH, sBaseE, sTotH, sChH, sChE;

    ssel[tid] = -1;

    __builtin_prefetch(pd + tid * 64, 0, 0);   // global_prefetch_b8 warm-up of dilated plane

    // ---- Pass A: total hard count -> excess ----
    int myH = 0;
    for (int t = 0; t < 64; ++t) {
        int idx = t * 1024 + tid;
        int l = lab[idx]; int d = pd[idx];
        myH += (l == cls && d != cls) ? 1 : 0;
    }
    for (int off = 16; off > 0; off >>= 1) myH += __shfl_down(myH, off, 32);
    if (lane == 0) wsH[wid] = myH;
    __syncthreads();
    if (tid == 0) {
        int tot = 0;
        for (int i = 0; i < 32; ++i) tot += wsH[i];
        sTotH = tot; sBaseH = 0; sBaseE = 0;
    }
    __syncthreads();
    int excess = sTotH - HALFS; if (excess < 0) excess = 0;

    // ---- Pass B: stable raster-order ranks and scatter ----
    for (int t = 0; t < 64; ++t) {
        int idx = t * 1024 + tid;
        int l = lab[idx]; int d = pd[idx];
        bool inc = (l == cls);
        int h = (inc && d != cls) ? 1 : 0;
        int e = (inc && d == cls) ? 1 : 0;

        int vH = h, vE = e;                         // wave inclusive scans
        for (int off = 1; off < 32; off <<= 1) {
            int nH = __shfl_up(vH, off, 32);
            int nE = __shfl_up(vE, off, 32);
            if (lane >= off) { vH += nH; vE += nE; }
        }
        if (lane == 31) { wsH[wid] = vH; wsE[wid] = vE; }
        __syncthreads();
        if (tid == 0) {                              // exclusive scan of wave totals
            int aH = 0, aE = 0;
            for (int i = 0; i < 32; ++i) {
                int tH = wsH[i]; wsH[i] = aH; aH += tH;
                int tE = wsE[i]; wsE[i] = aE; aE += tE;
            }
            sChH = aH; sChE = aE;
        }
        __syncthreads();
        int rH = sBaseH + wsH[wid] + (vH - h);       // global exclusive hard rank
        int rE = sBaseE + wsE[wid] + (vE - e);       // global exclusive easy rank
        // hard: slots [0,512) = hard_top; slots [512,1024) = excess hard prefix of rest
        if (h && rH < 1024) ssel[rH] = idx;
        // easy: rest slot = excess + easy_rank while < 512
        if (e && (excess + rE) < HALFS) ssel[HALFS + excess + rE] = idx;
        __syncthreads();
        if (tid == 0) { sBaseH += sChH; sBaseE += sChE; }
        __syncthreads();
    }
    s[tid] = ssel[tid];
}

// ---------------- Kernel 4: feature column gather + labels ----------------
__global__ void gather_kernel(const float* __restrict__ feat, const int* __restrict__ sel,
                              float* __restrict__ out) {
    int gid = blockIdx.x * blockDim.x + threadIdx.x;
    if (gid < PAIRS) out[OUT_FEAT + gid] = (float)(gid % 3 + 1);   // feat_label
    if (gid >= OUT_FEAT) return;
    int j = gid & 1023;
    int c = (gid >> 10) & 63;
    int p = gid >> 16;
    int b = p / 3;
    int idx = sel[p * 1024 + j];
    float v = 0.0f;
    if (idx >= 0) v = feat[((size_t)(b * CCH + c)) * N + idx];
    out[gid] = v;   // coalesced writes; gather reads are ascending within segments
}

extern "C" void kernel_launch(void* const* d_in, const int* in_sizes, int n_in,
                              void* d_out, int out_size, void* d_ws, size_t ws_size,
                              hipStream_t stream) {
    (void)in_sizes; (void)n_in; (void)out_size; (void)ws_size;
    const float* feat   = (const float*)d_in[0];
    const int*   labels = (const int*)d_in[1];
    const float* preds  = (const float*)d_in[2];
    float* out = (float*)d_out;

    uint8_t* ws  = (uint8_t*)d_ws;
    uint8_t* am  = ws;                               // B*N bytes
    uint8_t* dil = ws + (size_t)B * N;               // B*N bytes
    int*     sel = (int*)(ws + (size_t)2 * B * N);   // PAIRS*1024 ints (4B aligned)

    argmax_kernel<<<(B * N + 255) / 256, 256, 0, stream>>>(preds, am);
    dilate_kernel<<<dim3(W / 16, H / 16, B), dim3(16, 16), 0, stream>>>(am, dil);
    select_kernel<<<PAIRS, 1024, 0, stream>>>(labels, dil, sel);
    gather_kernel<<<(OUT_FEAT + 255) / 256, 256, 0, stream>>>(feat, sel, out);
}